// FusedSliceWhereReplacement_30872224924126
// MI455X (gfx1250) — compile-verified
//
#include <hip/hip_runtime.h>
#include <stdint.h>

// FusedSliceWhere: out[n][b][l] = where[b][l] ? slice[b][start[n]+l] : 0
// B=2048, T=8192, L=1024, N=64  (fixed by the reference)
//
// Memory-movement kernel: 512 MiB NT stores, slice row staged HBM->LDS once
// per b via async global->LDS loads, mask held in registers across the n loop.

typedef float v4f __attribute__((ext_vector_type(4)));
typedef int   gvec4 __attribute__((vector_size(4 * sizeof(int))));  // matches builtin param

#define AS1 __attribute__((address_space(1)))
#define AS3 __attribute__((address_space(3)))

constexpr int kB = 2048;
constexpr int kT = 8192;
constexpr int kL = 1024;
constexpr int kN = 64;
constexpr int kThreads = 256;

static_assert(kL == 4 * kThreads, "one block covers exactly one L-row in float4s");
static_assert(kT % (4 * kThreads) == 0, "row staging loop is exact");

__global__ __launch_bounds__(kThreads)
void fused_slice_where_kernel(const unsigned char* __restrict__ mask,   // [B, L] bool (1B each)
                              const float*         __restrict__ slice,  // [B, T]
                              const int*           __restrict__ starts, // [N]
                              float*               __restrict__ out)    // [N, B, L]
{
    __shared__ float row[kT];        // 32 KB: one full slice row
    __shared__ int   sstart[kN];     // 256 B: the 64 start offsets

    const int tid = threadIdx.x;
    const int b   = blockIdx.x;

    const float* gsrc = slice + (size_t)b * kT;

    // ---- Stage slice row HBM -> LDS (8 x b128 per thread) ----
#if __has_builtin(__builtin_amdgcn_global_load_async_to_lds_b128)
    #pragma unroll
    for (int i = 0; i < kT / (4 * kThreads); ++i) {
        const int idx = (i * kThreads + tid) * 4;
        __builtin_amdgcn_global_load_async_to_lds_b128(
            (AS1 gvec4*)(gsrc + idx),
            (AS3 gvec4*)(row + idx),
            /*offset=*/0, /*cpol=*/0);
    }
  #if __has_builtin(__builtin_amdgcn_s_wait_asynccnt)
    __builtin_amdgcn_s_wait_asynccnt(0);
  #else
    asm volatile("s_wait_asynccnt 0" ::: "memory");
  #endif
#else
    #pragma unroll
    for (int i = 0; i < kT / (4 * kThreads); ++i) {
        const int idx = (i * kThreads + tid) * 4;
        *(v4f*)(row + idx) = *(const v4f*)(gsrc + idx);
    }
#endif

    if (tid < kN) sstart[tid] = starts[tid];

    // ---- Mask: 4 bytes per thread, loaded once, live in registers for all n ----
    const unsigned mw = *(const unsigned*)(mask + (size_t)b * kL + tid * 4);
    const bool m0 = (mw & 0x000000ffu) != 0u;
    const bool m1 = (mw & 0x0000ff00u) != 0u;
    const bool m2 = (mw & 0x00ff0000u) != 0u;
    const bool m3 = (mw & 0xff000000u) != 0u;

    __syncthreads();

    float*       obase   = out + (size_t)b * kL + tid * 4;
    const size_t nstride = (size_t)kB * kL;

    #pragma unroll 4
    for (int n = 0; n < kN; ++n) {
        const int s = sstart[n];                 // uniform per wave
        const float* p = row + s + tid * 4;      // LDS, 4B-aligned (s arbitrary)
        v4f v;
        v.x = m0 ? p[0] : 0.0f;
        v.y = m1 ? p[1] : 0.0f;
        v.z = m2 ? p[2] : 0.0f;
        v.w = m3 ? p[3] : 0.0f;
        // 16B-aligned non-temporal store: don't pollute L2 with the 512 MiB output
        __builtin_nontemporal_store(v, (v4f*)(obase + (size_t)n * nstride));
    }
}

extern "C" void kernel_launch(void* const* d_in, const int* in_sizes, int n_in,
                              void* d_out, int out_size, void* d_ws, size_t ws_size,
                              hipStream_t stream) {
    // setup_inputs() order: where_input(bool[B,L]), slice_input(f32[B,T]),
    //                       slice_len(int scalar, unused), start_indices(i32[N])
    const unsigned char* mask   = (const unsigned char*)d_in[0];
    const float*         slice  = (const float*)d_in[1];
    const int*           starts = (const int*)d_in[3];
    float*               out    = (float*)d_out;

    fused_slice_where_kernel<<<kB, kThreads, 0, stream>>>(mask, slice, starts, out);
}